// UResNet_28759101014596
// MI455X (gfx1250) — compile-verified
//
#include <hip/hip_runtime.h>
#include <algorithm>

// ---------------------------------------------------------------------------
// Sparse UResNet forward for MI455X (gfx1250, wave32, WMMA).
// Core op: out[N,Cout] = sum_o gather(X, idx[o]) @ W[o], fp32 end-to-end via
// V_WMMA_F32_16X16X4_F32. Working set is L2-resident (192MB L2), so the
// target is instruction count per WMMA:
//  - weights pre-transposed to (o,cout,cin): A and B frags are contiguous
//    float2 loads
//  - invalid gather rows redirect the row POINTER to a zero row (no EXEC
//    churn, no per-load branches)
//  - one wave computes 2 row tiles x all Cout/16 col tiles: B frags shared
//    across row tiles, A frags shared across col tiles
//  - full tiles take an unguarded store path
// ---------------------------------------------------------------------------

#define NSTR 5
static const int NP[NSTR] = {16, 32, 48, 64, 80};
#define BN_EPS 1e-4f
#define BN_P 256       // fixed partial count -> deterministic BN reduction
#define CMAX 128       // max channel count (decoder concat at level 3)

typedef float v2f __attribute__((ext_vector_type(2)));
typedef float v8f __attribute__((ext_vector_type(8)));

// ---------------------------------------------------------------------------
// Templated gather-conv, fp32 WMMA 16x16x4.
//   X    : (Nin, CIN) fp32 row-major
//   IDX  : (NO, N) int32 gather rows (-1 => zero row), or nullptr => identity
//   Wt   : (NO, COUT, CIN) fp32 (pre-transposed)
//   zrow : >= CIN zero floats (invalid-gather target)
//   OUT  : (N, COUT) fp32, fully overwritten
// One wave: 32 rows (2 tiles) x COUT cols. A-frag (ISA 7.12.2, fp32 A 16x4):
// lane<16 -> M=lane, K={k,k+1}; lane>=16 -> M=lane-16, K={k+2,k+3}.
// B 4x16: v0=K(klo), v1=K(klo+1), N=lane%16. C/D: vgpr v -> M=v(+8 hi).
// ---------------------------------------------------------------------------
template <int CIN, int COUT>
__launch_bounds__(256)
__global__ void gconv_wmma_t(const float* __restrict__ X,
                             const int*   __restrict__ IDX,
                             const float* __restrict__ Wt,
                             const float* __restrict__ zrow,
                             float*       __restrict__ OUT,
                             int N, int NO)
{
    constexpr int NT = COUT / 16;
    const int lane = threadIdx.x & 31;
    const int wv   = threadIdx.x >> 5;
    const int tile = blockIdx.x * (blockDim.x >> 5) + wv;
    const int r0   = tile * 32;
    if (r0 >= N) return;                       // wave-uniform
    const int  m    = lane & 15;
    const bool hi   = lane >= 16;
    const int  kofs = hi ? 2 : 0;

    // clamped row ids: tail lanes gather a valid (unused) row, never stored
    const int row0  = r0 + m;
    const int row1  = r0 + 16 + m;
    const int rowc0 = row0 < N ? row0 : N - 1;
    const int rowc1 = row1 < N ? row1 : N - 1;

    v8f acc0[NT], acc1[NT];
#pragma unroll
    for (int t = 0; t < NT; ++t) { v8f z = {}; acc0[t] = z; acc1[t] = z; }

    for (int o = 0; o < NO; ++o) {
        const int ridx0 = IDX ? IDX[(size_t)o * N + rowc0] : rowc0;
        const int ridx1 = IDX ? IDX[(size_t)o * N + rowc1] : rowc1;
        // pointer redirect instead of per-load predication
        const float* x0 = ((ridx0 >= 0) ? (X + (size_t)ridx0 * CIN) : zrow) + kofs;
        const float* x1 = ((ridx1 >= 0) ? (X + (size_t)ridx1 * CIN) : zrow) + kofs;
        const float* wo = Wt + (size_t)o * COUT * CIN + (size_t)m * CIN + kofs;
#pragma unroll 4
        for (int k0 = 0; k0 < CIN; k0 += 4) {
            const v2f a0 = *(const v2f*)(x0 + k0);
            const v2f a1 = *(const v2f*)(x1 + k0);
#pragma unroll
            for (int t = 0; t < NT; ++t) {
                const v2f b = *(const v2f*)(wo + (size_t)t * 16 * CIN + k0);
                acc0[t] = __builtin_amdgcn_wmma_f32_16x16x4_f32(
                    false, a0, false, b, (short)0, acc0[t], false, false);
                acc1[t] = __builtin_amdgcn_wmma_f32_16x16x4_f32(
                    false, a1, false, b, (short)0, acc1[t], false, false);
            }
        }
    }

    const int mrow = r0 + (hi ? 8 : 0);
    if (r0 + 32 <= N) {                        // full 2-tile fast path
#pragma unroll
        for (int t = 0; t < NT; ++t) {
            const int col = t * 16 + m;
#pragma unroll
            for (int v = 0; v < 8; ++v) {
                OUT[(size_t)(mrow + v) * COUT + col]      = acc0[t][v];
                OUT[(size_t)(mrow + 16 + v) * COUT + col] = acc1[t][v];
            }
        }
    } else {
#pragma unroll
        for (int t = 0; t < NT; ++t) {
            const int col = t * 16 + m;
#pragma unroll
            for (int v = 0; v < 8; ++v) {
                const int ra = mrow + v;
                const int rb = mrow + 16 + v;
                if (ra < N) OUT[(size_t)ra * COUT + col] = acc0[t][v];
                if (rb < N) OUT[(size_t)rb * COUT + col] = acc1[t][v];
            }
        }
    }
}

// Runtime-dim fallback (same Wt layout); blockIdx.y = 16-col tile.
__launch_bounds__(256)
__global__ void gconv_generic(const float* __restrict__ X,
                              const int*   __restrict__ IDX,
                              const float* __restrict__ Wt,
                              const float* __restrict__ zrow,
                              float*       __restrict__ OUT,
                              int N, int Cin, int Cout, int NO)
{
    const int lane = threadIdx.x & 31;
    const int wv   = threadIdx.x >> 5;
    const int tile = blockIdx.x * (blockDim.x >> 5) + wv;
    const int r0   = tile * 16;
    if (r0 >= N) return;
    const int  m    = lane & 15;
    const bool hi   = lane >= 16;
    const int  kofs = hi ? 2 : 0;
    const int  col  = blockIdx.y * 16 + m;
    const int  row  = r0 + m;
    const int  rowc = row < N ? row : N - 1;

    v8f acc = {};
    for (int o = 0; o < NO; ++o) {
        const int ridx = IDX ? IDX[(size_t)o * N + rowc] : rowc;
        const float* xr = ((ridx >= 0) ? (X + (size_t)ridx * Cin) : zrow) + kofs;
        const float* wr = Wt + ((size_t)o * Cout + col) * Cin + kofs;
        for (int k0 = 0; k0 < Cin; k0 += 4) {
            const v2f a = *(const v2f*)(xr + k0);
            const v2f b = *(const v2f*)(wr + k0);
            acc = __builtin_amdgcn_wmma_f32_16x16x4_f32(
                false, a, false, b, (short)0, acc, false, false);
        }
    }
#pragma unroll
    for (int v = 0; v < 8; ++v) {
        const int rr = r0 + v + (hi ? 8 : 0);
        if (rr < N) OUT[(size_t)rr * Cout + col] = acc[v];
    }
}

// Wt[o][c][k] = W[o][k][c]
__global__ void transpose_w(const float* __restrict__ W, float* __restrict__ Wt,
                            int NO, int Cin, int Cout)
{
    const size_t total = (size_t)NO * Cin * Cout;
    size_t i = (size_t)blockIdx.x * blockDim.x + threadIdx.x;
    const size_t stride = (size_t)gridDim.x * blockDim.x;
    for (; i < total; i += stride) {
        const int    k   = (int)(i % Cin);
        const size_t rem = i / Cin;
        const int    c   = (int)(rem % Cout);
        const int    o   = (int)(rem / Cout);
        Wt[i] = W[((size_t)o * Cin + k) * Cout + c];
    }
}

__global__ void zero_fill(float* __restrict__ p, int n)
{
    const int i = blockIdx.x * blockDim.x + threadIdx.x;
    if (i < n) p[i] = 0.0f;
}

// ---------------------------------------------------------------------------
// Input SubmanifoldConvolution(1 -> 16): out[i][c] = sum_o f[nbr[o][i]]*w[o][c]
// ---------------------------------------------------------------------------
__global__ void conv_in_kernel(const float* __restrict__ pc,
                               const int*   __restrict__ nbr,
                               const float* __restrict__ w_in,
                               float*       __restrict__ out, int N)
{
    __shared__ float ws[27 * 16];
    for (int t = threadIdx.x; t < 27 * 16; t += blockDim.x) ws[t] = w_in[t];
    __syncthreads();
    const int i = blockIdx.x * blockDim.x + threadIdx.x;
    if (i >= N) return;
    float a[16];
#pragma unroll
    for (int c = 0; c < 16; ++c) a[c] = 0.0f;
    for (int o = 0; o < 27; ++o) {
        const int j  = nbr[(size_t)o * N + i];
        const int jc = j < 0 ? 0 : j;
        float f = pc[(size_t)jc * 5 + 4];      // feats = pc[:, -1]
        f = j < 0 ? 0.0f : f;                  // select, not branch
#pragma unroll
        for (int c = 0; c < 16; ++c) a[c] += f * ws[o * 16 + c];
    }
#pragma unroll
    for (int c = 0; c < 16; ++c) out[(size_t)i * 16 + c] = a[c];
}

// ---------------------------------------------------------------------------
// BN (training batch stats): deterministic 2-stage reduction + fused apply.
// blockDim == C everywhere; thread = channel; fixed accumulate order.
// ---------------------------------------------------------------------------
__global__ void bn_stats(const float* __restrict__ x, int N, int C,
                         float* __restrict__ partial)
{
    const int c = threadIdx.x;
    float s = 0.0f, s2 = 0.0f;
    for (int r = blockIdx.x; r < N; r += gridDim.x) {
        const float v = x[(size_t)r * C + c];
        s += v; s2 += v * v;
    }
    partial[(size_t)blockIdx.x * 2 * C + c]     = s;
    partial[(size_t)blockIdx.x * 2 * C + C + c] = s2;
}

__global__ void bn_finalize(const float* __restrict__ partial, int P, int N, int C,
                            const float* __restrict__ g, const float* __restrict__ b,
                            float* __restrict__ ss)
{
    const int c = threadIdx.x;
    float s = 0.0f, s2 = 0.0f;
    for (int p = 0; p < P; ++p) {
        s  += partial[(size_t)p * 2 * C + c];
        s2 += partial[(size_t)p * 2 * C + C + c];
    }
    const float inv = 1.0f / (float)N;
    const float mu  = s * inv;
    const float var = s2 * inv - mu * mu;
    const float sc  = g[c] * rsqrtf(var + BN_EPS);
    ss[c]     = sc;
    ss[C + c] = b[c] - mu * sc;
}

__global__ void bn_apply(const float* __restrict__ x, const float* __restrict__ ss,
                         float* __restrict__ y, int N, int C)
{
    const int c = threadIdx.x;                 // blockDim == C
    const float sc = ss[c];
    const float sh = ss[C + c];
    for (int r = blockIdx.x; r < N; r += gridDim.x) {
        const float v = x[(size_t)r * C + c] * sc + sh;
        y[(size_t)r * C + c] = v > 0.0f ? v : 0.0f;
    }
}

// out = a + b (elementwise; out may alias a)
__global__ void add2(const float* __restrict__ a, const float* __restrict__ b,
                     float* __restrict__ o, size_t total)
{
    size_t i = (size_t)blockIdx.x * blockDim.x + threadIdx.x;
    const size_t stride = (size_t)gridDim.x * blockDim.x;
    for (; i < total; i += stride) o[i] = a[i] + b[i];
}

// out[N, C1+C2] = [a | b]; blockDim == C1+C2
__global__ void concat2(const float* __restrict__ a, const float* __restrict__ b,
                        float* __restrict__ o, int N, int C1, int C2)
{
    const int c = threadIdx.x;
    const int C = C1 + C2;
    const bool first = c < C1;
    const int  cc    = first ? c : c - C1;
    for (int r = blockIdx.x; r < N; r += gridDim.x)
        o[(size_t)r * C + c] = first ? a[(size_t)r * C1 + cc]
                                     : b[(size_t)r * C2 + cc];
}

// Deconv -> gather-conv rulebook: idx[o][i] = (off[i]==o) ? parent[i] : -1
__global__ void build_didx(const int* __restrict__ parent, const int* __restrict__ off,
                           int* __restrict__ idx, int N)
{
    const int i = blockIdx.x * blockDim.x + threadIdx.x;
    if (i >= N) return;
    const int p = parent[i];
    const int f = off[i];
#pragma unroll
    for (int o = 0; o < 8; ++o) idx[(size_t)o * N + i] = (f == o) ? p : -1;
}

// Head: out[N,5] = x[N,16] @ w[16,5] + b[5]
__global__ void linear_head(const float* __restrict__ x, const float* __restrict__ w,
                            const float* __restrict__ b, float* __restrict__ out, int N)
{
    __shared__ float sw[16 * 5];
    __shared__ float sb[5];
    for (int t = threadIdx.x; t < 80; t += blockDim.x) sw[t] = w[t];
    if (threadIdx.x < 5) sb[threadIdx.x] = b[threadIdx.x];
    __syncthreads();
    const int i = blockIdx.x * blockDim.x + threadIdx.x;
    if (i >= N) return;
    float xi[16];
#pragma unroll
    for (int c = 0; c < 16; ++c) xi[c] = x[(size_t)i * 16 + c];
#pragma unroll
    for (int k = 0; k < 5; ++k) {
        float acc = sb[k];
#pragma unroll
        for (int c = 0; c < 16; ++c) acc += xi[c] * sw[c * 5 + k];
        out[(size_t)i * 5 + k] = acc;
    }
}

// ---------------------------------------------------------------------------
// Host orchestration. Input flattening assumption: recursive insertion-order
// flatten of the setup_inputs() dict (159 tensors total):
//   pc, [w_in, enc(5x2x{g1,b1,w1,g2,b2,w2}), enc_down(4x{g,b,w}),
//   dec_up(4x{g,b,w}), dec(4x{blk0:{g1,b1,w1,g2,b2,w2,nin}, blk1:{6}}),
//   out_g, out_b, lin_w, lin_b], nbr[5], down[4], up_parent[4], up_off[4]
// ---------------------------------------------------------------------------
struct Blk { const float *g1, *b1, *w1, *g2, *b2, *w2, *nin;
             float *w1t, *w2t, *nint; };
struct BnW { const float *g, *b, *w; float* wt; };
struct TJob { const float* w; float* wt; int no, ci, co; };

extern "C" void kernel_launch(void* const* d_in, const int* in_sizes, int n_in,
                              void* d_out, int out_size, void* d_ws, size_t ws_size,
                              hipStream_t stream)
{
    (void)n_in; (void)out_size; (void)ws_size;
    int cur = 0;
    auto F = [&]() { return (const float*)d_in[cur++]; };
    auto I = [&]() { return (const int*)d_in[cur++]; };

    const float* pc = F();
    const long N0 = in_sizes[0] / 5;
    const float* w_in = F();

    Blk enc[NSTR][2];
    for (int i = 0; i < NSTR; ++i)
        for (int b = 0; b < 2; ++b) {
            enc[i][b].g1 = F(); enc[i][b].b1 = F(); enc[i][b].w1 = F();
            enc[i][b].g2 = F(); enc[i][b].b2 = F(); enc[i][b].w2 = F();
            enc[i][b].nin = nullptr; enc[i][b].nint = nullptr;
        }
    BnW encd[4];
    for (int i = 0; i < 4; ++i) { encd[i].g = F(); encd[i].b = F(); encd[i].w = F(); }
    BnW decu[4];
    for (int j = 0; j < 4; ++j) { decu[j].g = F(); decu[j].b = F(); decu[j].w = F(); }
    Blk dec[4][2];
    for (int j = 0; j < 4; ++j) {
        dec[j][0].g1 = F(); dec[j][0].b1 = F(); dec[j][0].w1 = F();
        dec[j][0].g2 = F(); dec[j][0].b2 = F(); dec[j][0].w2 = F();
        dec[j][0].nin = F();                                   // a != b -> nin last
        dec[j][1].g1 = F(); dec[j][1].b1 = F(); dec[j][1].w1 = F();
        dec[j][1].g2 = F(); dec[j][1].b2 = F(); dec[j][1].w2 = F();
        dec[j][1].nin = nullptr; dec[j][1].nint = nullptr;
    }
    const float* out_g = F();
    const float* out_b = F();
    const float* lin_w = F();
    const float* lin_b = F();

    const int nbr_base = cur;
    const int* nbr[5];  for (int l = 0; l < 5; ++l) nbr[l]  = I();
    const int* down[4]; for (int l = 0; l < 4; ++l) down[l] = I();
    const int* upp[4];  for (int l = 0; l < 4; ++l) upp[l]  = I();
    const int* upo[4];  for (int l = 0; l < 4; ++l) upo[l]  = I();

    long Nl[5];
    Nl[0] = N0;
    for (int l = 1; l < 5; ++l) Nl[l] = in_sizes[nbr_base + l] / 27;

    // ---- workspace carve (runtime-sized, 256B aligned) ----
    size_t off = 0;
    auto alloc = [&](size_t bytes) -> char* {
        char* p = (char*)d_ws + off;
        off += (bytes + 255) & ~(size_t)255;
        return p;
    };
    float* ssBuf  = (float*)alloc(2 * CMAX * sizeof(float));
    float* zrow   = (float*)alloc(CMAX * sizeof(float));
    float* bnPart = (float*)alloc((size_t)BN_P * 2 * CMAX * sizeof(float));
    float* skip[5];
    for (int l = 0; l < 5; ++l)
        skip[l] = (float*)alloc((size_t)Nl[l] * NP[l] * sizeof(float));
    int* didx = (int*)alloc((size_t)8 * Nl[0] * sizeof(int));
    size_t cap2 = 0, cap1 = 0;
    for (int l = 0; l < 5; ++l) {
        cap2 = std::max(cap2, (size_t)Nl[l] * 2 * NP[l]);
        cap1 = std::max(cap1, (size_t)Nl[l] * NP[l]);
    }
    float* bufA = (float*)alloc(cap2 * sizeof(float));  // current x (<= 2*NP wide)
    float* bufB = (float*)alloc(cap2 * sizeof(float));  // BN-ReLU output
    float* bufC = (float*)alloc(cap1 * sizeof(float));  // conv output h
    float* bufD = (float*)alloc(cap1 * sizeof(float));  // shortcut / deconv out

    // ---- weight transpose slots (Wt layout: (NO, Cout, Cin)) ----
    TJob jobs[64]; int nj = 0;
    auto addT = [&](const float* w, int no, int ci, int co) -> float* {
        float* wt = (float*)alloc((size_t)no * ci * co * sizeof(float));
        jobs[nj].w = w; jobs[nj].wt = wt; jobs[nj].no = no;
        jobs[nj].ci = ci; jobs[nj].co = co; ++nj;
        return wt;
    };
    for (int i = 0; i < NSTR; ++i)
        for (int b = 0; b < 2; ++b) {
            enc[i][b].w1t = addT(enc[i][b].w1, 27, NP[i], NP[i]);
            enc[i][b].w2t = addT(enc[i][b].w2, 27, NP[i], NP[i]);
        }
    for (int i = 0; i < 4; ++i) encd[i].wt = addT(encd[i].w, 8, NP[i], NP[i + 1]);
    for (int j = 0; j < 4; ++j) {
        const int lvl = 3 - j;
        decu[j].wt = addT(decu[j].w, 8, NP[lvl + 1], NP[lvl]);
        dec[j][0].w1t  = addT(dec[j][0].w1, 27, 2 * NP[lvl], NP[lvl]);
        dec[j][0].w2t  = addT(dec[j][0].w2, 27, NP[lvl], NP[lvl]);
        dec[j][0].nint = addT(dec[j][0].nin, 1, 2 * NP[lvl], NP[lvl]);
        dec[j][1].w1t  = addT(dec[j][1].w1, 27, NP[lvl], NP[lvl]);
        dec[j][1].w2t  = addT(dec[j][1].w2, 27, NP[lvl], NP[lvl]);
    }
    zero_fill<<<1, CMAX, 0, stream>>>(zrow, CMAX);
    for (int t = 0; t < nj; ++t) {
        const size_t total = (size_t)jobs[t].no * jobs[t].ci * jobs[t].co;
        const int blocks = (int)std::min<size_t>((total + 255) / 256, 512);
        transpose_w<<<blocks, 256, 0, stream>>>(jobs[t].w, jobs[t].wt,
                                                jobs[t].no, jobs[t].ci, jobs[t].co);
    }

    // ---- launch helpers ----
    auto run_bn = [&](const float* x, long Nn, int C, const float* g,
                      const float* b, float* y) {
        bn_stats<<<BN_P, C, 0, stream>>>(x, (int)Nn, C, bnPart);
        bn_finalize<<<1, C, 0, stream>>>(bnPart, BN_P, (int)Nn, C, g, b, ssBuf);
        const int blocks = (int)std::min<long>(Nn, 2048);
        bn_apply<<<blocks, C, 0, stream>>>(x, ssBuf, y, (int)Nn, C);
    };
    auto run_conv = [&](const float* X, const int* IDX, const float* Wt, float* O,
                        long Nn, int Cin, int Cout, int NO) {
        const long tiles = (Nn + 31) / 32;
        dim3 grid((unsigned)((tiles + 7) / 8));
#define GCONV_CASE(ci, co)                                                       \
        if (Cin == ci && Cout == co) {                                           \
            gconv_wmma_t<ci, co><<<grid, 256, 0, stream>>>(X, IDX, Wt, zrow, O,  \
                                                           (int)Nn, NO);         \
            return;                                                              \
        }
        GCONV_CASE(16, 16)  GCONV_CASE(32, 16)
        GCONV_CASE(16, 32)  GCONV_CASE(32, 32)  GCONV_CASE(48, 32) GCONV_CASE(64, 32)
        GCONV_CASE(32, 48)  GCONV_CASE(48, 48)  GCONV_CASE(64, 48) GCONV_CASE(96, 48)
        GCONV_CASE(48, 64)  GCONV_CASE(64, 64)  GCONV_CASE(80, 64) GCONV_CASE(128, 64)
        GCONV_CASE(64, 80)  GCONV_CASE(80, 80)
#undef GCONV_CASE
        const long tiles16 = (Nn + 15) / 16;
        dim3 grid2((unsigned)((tiles16 + 7) / 8), (unsigned)(Cout / 16));
        gconv_generic<<<grid2, 256, 0, stream>>>(X, IDX, Wt, zrow, O,
                                                 (int)Nn, Cin, Cout, NO);
    };
    auto run_add = [&](const float* a, const float* b, float* o, size_t total) {
        const int blocks = (int)std::min<size_t>((total + 255) / 256, 2048);
        add2<<<blocks, 256, 0, stream>>>(a, b, o, total);
    };
    auto run_block = [&](const Blk& p, const int* nb, long Nn, int Cin, int Cout) {
        // x lives in bufA (Nn x Cin); result returned in bufA (Nn x Cout)
        const float* shortcut;
        if (p.nint) { run_conv(bufA, nullptr, p.nint, bufD, Nn, Cin, Cout, 1); shortcut = bufD; }
        else shortcut = bufA;
        run_bn(bufA, Nn, Cin, p.g1, p.b1, bufB);
        run_conv(bufB, nb, p.w1t, bufC, Nn, Cin, Cout, 27);
        run_bn(bufC, Nn, Cout, p.g2, p.b2, bufB);
        run_conv(bufB, nb, p.w2t, bufC, Nn, Cout, Cout, 27);
        run_add(shortcut, bufC, bufA, (size_t)Nn * Cout);
    };

    // ---- forward ----
    conv_in_kernel<<<(int)((Nl[0] + 255) / 256), 256, 0, stream>>>(
        pc, nbr[0], w_in, bufA, (int)Nl[0]);

    // encoder
    for (int i = 0; i < NSTR; ++i) {
        for (int b = 0; b < 2; ++b) run_block(enc[i][b], nbr[i], Nl[i], NP[i], NP[i]);
        hipMemcpyAsync(skip[i], bufA, (size_t)Nl[i] * NP[i] * sizeof(float),
                       hipMemcpyDeviceToDevice, stream);
        if (i < NSTR - 1) {
            run_bn(bufA, Nl[i], NP[i], encd[i].g, encd[i].b, bufB);
            run_conv(bufB, down[i], encd[i].wt, bufA, Nl[i + 1], NP[i], NP[i + 1], 8);
        }
    }

    // decoder
    for (int j = 0; j < 4; ++j) {
        const int lvl = 3 - j;
        run_bn(bufA, Nl[lvl + 1], NP[lvl + 1], decu[j].g, decu[j].b, bufB);
        build_didx<<<(int)((Nl[lvl] + 255) / 256), 256, 0, stream>>>(
            upp[lvl], upo[lvl], didx, (int)Nl[lvl]);
        run_conv(bufB, didx, decu[j].wt, bufD, Nl[lvl], NP[lvl + 1], NP[lvl], 8);
        {
            const int blocks = (int)std::min<long>(Nl[lvl], 2048);
            concat2<<<blocks, 2 * NP[lvl], 0, stream>>>(skip[lvl], bufD, bufA,
                                                        (int)Nl[lvl], NP[lvl], NP[lvl]);
        }
        run_block(dec[j][0], nbr[lvl], Nl[lvl], 2 * NP[lvl], NP[lvl]);
        run_block(dec[j][1], nbr[lvl], Nl[lvl], NP[lvl], NP[lvl]);
    }

    // head
    run_bn(bufA, Nl[0], 16, out_g, out_b, bufB);
    linear_head<<<(int)((Nl[0] + 255) / 256), 256, 0, stream>>>(
        bufB, lin_w, lin_b, (float*)d_out, (int)Nl[0]);
}